// UVIN_84851373899879
// MI455X (gfx1250) — compile-verified
//
#include <hip/hip_runtime.h>

#define IMSIZE 64
#define SSTATES (IMSIZE * IMSIZE)   // 4096
#define A_SIZE 8
#define C_SIZE 10
#define AC (A_SIZE * C_SIZE)        // 80
#define VI_K 30
#define GAMMA 0.99f
#define BATCH 32
#define SB 128
#define VI_THREADS 1024
#define SPT (SSTATES / VI_THREADS)  // 4 states per thread

typedef __attribute__((ext_vector_type(2))) float v2f;
typedef __attribute__((ext_vector_type(8))) float v8f;

// ---------------------------------------------------------------------------
// Kernel 0: cp[i] = clip(pv[ds_prob[i]], 0, 1)  -- kills one indirection level
// in the VI inner loop (327,680 elements, stays hot in L2 afterwards).
// ---------------------------------------------------------------------------
__global__ void prep_cp_kernel(const int* __restrict__ ds_prob,
                               const float* __restrict__ pv,
                               float* __restrict__ cp, int n) {
  int i = blockIdx.x * blockDim.x + threadIdx.x;
  if (i < n) {
    float p = pv[ds_prob[i]];
    cp[i] = fminf(fmaxf(p, 0.0f), 1.0f);
  }
}

// ---------------------------------------------------------------------------
// Kernel 1: fused conv + 30 value-iteration sweeps. One workgroup per batch
// element; v[b,:] (16 KB), sar (16 KB), g = GAMMA*(1-sad) (16 KB) live in LDS.
// Each thread owns 4 states and all 8 actions of each -> the max over actions
// is thread-local, so each iteration needs only two barriers.
// ---------------------------------------------------------------------------
__global__ __launch_bounds__(VI_THREADS)
void vi_kernel(const float* __restrict__ x,
               const int* __restrict__ cf,      // ds_state, [S][8][10]
               const float* __restrict__ cp,    // precomputed, [S][8][10]
               const float* __restrict__ conv_w,
               const float* __restrict__ conv_b,
               float* __restrict__ qfull) {     // [B][S][8]
  __shared__ float v_lds[SSTATES];
  __shared__ float sar_lds[SSTATES];
  __shared__ float g_lds[SSTATES];

  const int b = blockIdx.x;
  const int t = threadIdx.x;
  const float* x0 = x + (size_t)b * 2 * SSTATES;
  const float* x1 = x0 + SSTATES;

  float w0[9], w1[9];
#pragma unroll
  for (int k = 0; k < 9; ++k) { w0[k] = conv_w[k]; w1[k] = conv_w[9 + k]; }
  const float cb = conv_b[0];

  // conv (SAME, correlation) + sad/g + v0 = sar
#pragma unroll
  for (int i = 0; i < SPT; ++i) {
    int s = t + i * VI_THREADS;
    int ii = s >> 6, jj = s & 63;
    float acc = cb;
#pragma unroll
    for (int ky = 0; ky < 3; ++ky) {
      int yy = ii + ky - 1;
#pragma unroll
      for (int kx = 0; kx < 3; ++kx) {
        int xx = jj + kx - 1;
        bool ok = ((unsigned)yy < (unsigned)IMSIZE) & ((unsigned)xx < (unsigned)IMSIZE);
        float a0 = ok ? x0[yy * IMSIZE + xx] : 0.0f;
        float a1 = ok ? x1[yy * IMSIZE + xx] : 0.0f;
        acc = fmaf(a0, w0[ky * 3 + kx], acc);
        acc = fmaf(a1, w1[ky * 3 + kx], acc);
      }
    }
    sar_lds[s] = acc;
    v_lds[s]   = acc;                            // v0 = sar
    g_lds[s]   = GAMMA * (1.0f - x1[s] * 0.1f);  // GAMMA*(1 - sad)
  }
  __syncthreads();

  float q[SPT][A_SIZE];
  for (int k = 0; k < VI_K; ++k) {
    float vnew[SPT];
#pragma unroll
    for (int i = 0; i < SPT; ++i) {
      int s = t + i * VI_THREADS;
      float sar = sar_lds[s];
      float g   = g_lds[s];
      const int*   cfp = cf + (size_t)s * AC;
      const float* cpp = cp + (size_t)s * AC;
      float vm = -3.4e38f;
#pragma unroll
      for (int a = 0; a < A_SIZE; ++a) {
        float acc = 0.0f;
#pragma unroll
        for (int c = 0; c < C_SIZE; ++c) {
          acc = fmaf(cpp[a * C_SIZE + c], v_lds[cfp[a * C_SIZE + c]], acc);
        }
        float qa = fmaf(g, acc, sar);
        q[i][a] = qa;
        vm = fmaxf(vm, qa);
      }
      vnew[i] = vm;
    }
    __syncthreads();
#pragma unroll
    for (int i = 0; i < SPT; ++i) v_lds[t + i * VI_THREADS] = vnew[i];
    __syncthreads();
  }

  // dump last-iteration q to workspace
  float* qb = qfull + (size_t)b * SSTATES * A_SIZE;
#pragma unroll
  for (int i = 0; i < SPT; ++i) {
    int s = t + i * VI_THREADS;
#pragma unroll
    for (int a = 0; a < A_SIZE; ++a) qb[s * A_SIZE + a] = q[i][a];
  }
}

// ---------------------------------------------------------------------------
// Kernel 2: gather (flat = s1*64+s2) + 8x8 linear w/ residual via
// V_WMMA_F32_16X16X4_F32. One wave per 16 output rows; K=8 done as two
// K=4 WMMA steps accumulating into C. fp32 end-to-end.
// ---------------------------------------------------------------------------
__global__ __launch_bounds__(256)
void lin_gather_wmma_kernel(const float* __restrict__ qfull,  // [B][S][8]
                            const int* __restrict__ s1,
                            const int* __restrict__ s2,
                            const float* __restrict__ lin_w,  // [8][8]
                            const float* __restrict__ lin_b,  // [8]
                            float* __restrict__ out) {        // [B*SB][8]
  const int gtid = blockIdx.x * blockDim.x + threadIdx.x;
  const int wave = gtid >> 5;           // wave32
  const int lane = threadIdx.x & 31;
  const int row0 = wave * 16;           // 16 output rows per wave
  const int n    = lane & 15;
  const int kb   = (lane >> 4) * 2;     // K sub-offset: 0 or 2

  // --- A fragment: gathered q rows. lane<16 -> M=lane,K={0,1}; lane>=16 -> K={2,3}
  int arow = row0 + n;
  int bidx = arow >> 7;                 // / SB
  int iidx = arow & (SB - 1);
  int flat = s1[bidx * SB + iidx] * IMSIZE + s2[bidx * SB + iidx];
  const float* ap = qfull + ((size_t)bidx * SSTATES + flat) * A_SIZE;
  v2f a0, a1;
  a0.x = ap[kb + 0];     a0.y = ap[kb + 1];       // K block 0..3
  a1.x = ap[4 + kb + 0]; a1.y = ap[4 + kb + 1];   // K block 4..7

  // --- B fragment: Bm[k][n] = lin_w[n][k] (i.e. lin_w^T), zero for n>=8.
  // Mask via multiply so EXEC stays all-ones around the WMMAs.
  float mask = (n < A_SIZE) ? 1.0f : 0.0f;
  int nn = n & (A_SIZE - 1);
  v2f b0, b1;
  b0.x = mask * lin_w[nn * A_SIZE + kb + 0];
  b0.y = mask * lin_w[nn * A_SIZE + kb + 1];
  b1.x = mask * lin_w[nn * A_SIZE + 4 + kb + 0];
  b1.y = mask * lin_w[nn * A_SIZE + 4 + kb + 1];

  v8f c = {};
  c = __builtin_amdgcn_wmma_f32_16x16x4_f32(false, a0, false, b0, (short)0, c, false, false);
  c = __builtin_amdgcn_wmma_f32_16x16x4_f32(false, a1, false, b1, (short)0, c, false, false);

  // --- epilogue: D + bias + residual(q). VGPR r holds row0 + r + 8*(lane>=16), col n.
  int moff = (lane >> 4) * 8;
  if (n < A_SIZE) {
    float bias = lin_b[n];
#pragma unroll
    for (int r = 0; r < 8; ++r) {
      int row = row0 + r + moff;
      int rb = row >> 7, ri = row & (SB - 1);
      int fl = s1[rb * SB + ri] * IMSIZE + s2[rb * SB + ri];
      float qv = qfull[((size_t)rb * SSTATES + fl) * A_SIZE + n];
      out[(size_t)row * A_SIZE + n] = c[r] + bias + qv;
    }
  }
}

// ---------------------------------------------------------------------------
extern "C" void kernel_launch(void* const* d_in, const int* in_sizes, int n_in,
                              void* d_out, int out_size, void* d_ws, size_t ws_size,
                              hipStream_t stream) {
  const float* x       = (const float*)d_in[0];
  const int*   s1      = (const int*)d_in[1];
  const int*   s2      = (const int*)d_in[2];
  const int*   ds_state= (const int*)d_in[3];
  const int*   ds_prob = (const int*)d_in[4];
  const float* conv_w  = (const float*)d_in[5];
  const float* conv_b  = (const float*)d_in[6];
  const float* pv      = (const float*)d_in[7];
  const float* lin_w   = (const float*)d_in[8];
  const float* lin_b   = (const float*)d_in[9];

  // workspace layout: cp [S*80] f32, then qfull [B*S*8] f32  (~5.5 MB total)
  float* cp    = (float*)d_ws;
  float* qfull = cp + (size_t)SSTATES * AC;

  const int ncp = SSTATES * AC;  // 327,680
  prep_cp_kernel<<<(ncp + 255) / 256, 256, 0, stream>>>(ds_prob, pv, cp, ncp);

  vi_kernel<<<BATCH, VI_THREADS, 0, stream>>>(x, ds_state, cp, conv_w, conv_b, qfull);

  // 4096 rows total, 16 rows/wave, 8 waves/block -> 32 blocks
  const int nrows = BATCH * SB;
  lin_gather_wmma_kernel<<<nrows / (16 * 8), 256, 0, stream>>>(
      qfull, s1, s2, lin_w, lin_b, (float*)d_out);
}